// ENMF_78847009620485
// MI455X (gfx1250) — compile-verified
//
#include <hip/hip_runtime.h>

// ENMF loss for MI455X (gfx1250, wave32).
// Gram kernels use V_WMMA_F32_16X16X4_F32: 64x64 Gram = 4x4 tiles of 16x16 f32
// accumulators (128 VGPRs/wave), K streamed 4 rows/chunk, one 8-float load per
// lane per chunk. launch_bounds(128,1) keeps the 16 accumulator tiles resident
// (no scratch spills).

typedef __attribute__((ext_vector_type(2))) float v2f;
typedef __attribute__((ext_vector_type(8))) float v8f;

#define NEG_WEIGHT 0.5f
#define D 64
#define GRAM_BLOCK 128  // 4 waves -> 1 wave per SIMD32 in a WGP

__global__ void init_ws_kernel(float* ws, int n) {
    int i = blockIdx.x * blockDim.x + threadIdx.x;
    if (i < n) ws[i] = 0.f;
}

__device__ __forceinline__ void gram_zero(v8f c[4][4]) {
#pragma unroll
    for (int i = 0; i < 4; ++i)
#pragma unroll
        for (int j = 0; j < 4; ++j)
            c[i][j] = (v8f){0.f, 0.f, 0.f, 0.f, 0.f, 0.f, 0.f, 0.f};
}

__device__ __forceinline__ void gram_mma(v8f c[4][4], const v2f op[4]) {
#pragma unroll
    for (int i = 0; i < 4; ++i)
#pragma unroll
        for (int j = 0; j < 4; ++j)
            c[i][j] = __builtin_amdgcn_wmma_f32_16x16x4_f32(
                false, op[i], false, op[j], (short)0, c[i][j], false, false);
}

// C/D layout: VGPR r holds (M=r, N=lane) lanes 0-15, (M=r+8, N=lane-16) lanes 16-31.
__device__ __forceinline__ void gram_reduce(v8f c[4][4], float* lds,
                                            float* __restrict__ acc,
                                            int col, int khalf) {
    const int rowAdd = khalf * 8;
#pragma unroll
    for (int i = 0; i < 4; ++i)
#pragma unroll
        for (int j = 0; j < 4; ++j)
#pragma unroll
            for (int r = 0; r < 8; ++r)
                atomicAdd(&lds[(i * 16 + r + rowAdd) * D + j * 16 + col],
                          c[i][j][r]);
    __syncthreads();
    for (int i = threadIdx.x; i < D * D; i += blockDim.x)
        atomicAdd(&acc[i], lds[i]);
}

// part_1: acc += W^T W over all rows (streams the 256 MB item table once).
__global__ __launch_bounds__(GRAM_BLOCK, 1)
void gram64_item_kernel(const float* __restrict__ W, int nrows,
                        float* __restrict__ acc) {
    __shared__ float lds[D * D];
    for (int i = threadIdx.x; i < D * D; i += blockDim.x) lds[i] = 0.f;
    __syncthreads();

    const int lane  = threadIdx.x & 31;
    const int col   = lane & 15;   // tile column (and row) index
    const int khalf = lane >> 4;   // 0 -> K={0,1}, 1 -> K={2,3}

    const int wavesPerBlock = GRAM_BLOCK >> 5;
    const int gwave  = blockIdx.x * wavesPerBlock + (threadIdx.x >> 5);
    const int nwaves = gridDim.x * wavesPerBlock;

    v8f c[4][4];
    gram_zero(c);

    const long long nFull  = nrows >> 2;                 // full 4-row chunks
    const long long stride = (long long)nwaves * 4 * D;  // floats per step
    const float* p = W + ((long long)gwave * 4 + 2 * khalf) * D + col;

    for (long long chunk = gwave; chunk < nFull; chunk += nwaves, p += stride) {
        if (chunk + nwaves < nFull) __builtin_prefetch(p + stride, 0, 0);
        // A-tile(i) and B-tile(j) per-lane operands coincide for A^T A:
        // op[g] = { W[r0, 16g+col], W[r0+1, 16g+col] }, r0 = chunk*4 + 2*khalf
        v2f op[4];
#pragma unroll
        for (int g = 0; g < 4; ++g)
            op[g] = (v2f){p[g * 16], p[D + g * 16]};
        gram_mma(c, op);
    }

    // remainder rows (nrows % 4), handled once by global wave 0
    const int rem = nrows & 3;
    if (rem && gwave == 0) {
        const long long r0 = nFull * 4 + 2 * khalf;
        const long long r1 = r0 + 1;
        v2f op[4];
#pragma unroll
        for (int g = 0; g < 4; ++g) {
            float a0 = (r0 < nrows) ? W[r0 * D + g * 16 + col] : 0.f;
            float a1 = (r1 < nrows) ? W[r1 * D + g * 16 + col] : 0.f;
            op[g] = (v2f){a0, a1};
        }
        gram_mma(c, op);
    }

    gram_reduce(c, lds, acc, col, khalf);
}

// part_2: acc += E^T E with E[n] = W[idx[n]] (gathered user embeddings).
__global__ __launch_bounds__(GRAM_BLOCK, 1)
void gram64_user_kernel(const float* __restrict__ W,
                        const int* __restrict__ idx, int nrows,
                        float* __restrict__ acc) {
    __shared__ float lds[D * D];
    for (int i = threadIdx.x; i < D * D; i += blockDim.x) lds[i] = 0.f;
    __syncthreads();

    const int lane  = threadIdx.x & 31;
    const int col   = lane & 15;
    const int khalf = lane >> 4;

    const int wavesPerBlock = GRAM_BLOCK >> 5;
    const int gwave  = blockIdx.x * wavesPerBlock + (threadIdx.x >> 5);
    const int nwaves = gridDim.x * wavesPerBlock;

    v8f c[4][4];
    gram_zero(c);

    const long long nFull = nrows >> 2;
    for (long long chunk = gwave; chunk < nFull; chunk += nwaves) {
        const long long r0 = chunk * 4 + 2 * khalf;
        const long long m0 = (long long)idx[r0];
        const long long m1 = (long long)idx[r0 + 1];
        const float* p0 = W + m0 * D + col;
        const float* p1 = W + m1 * D + col;
        v2f op[4];
#pragma unroll
        for (int g = 0; g < 4; ++g)
            op[g] = (v2f){p0[g * 16], p1[g * 16]};
        gram_mma(c, op);
    }

    const int rem = nrows & 3;
    if (rem && gwave == 0) {
        const long long r0 = nFull * 4 + 2 * khalf;
        const long long r1 = r0 + 1;
        v2f op[4];
#pragma unroll
        for (int g = 0; g < 4; ++g) {
            float a0 = (r0 < nrows) ? W[(long long)idx[r0] * D + g * 16 + col] : 0.f;
            float a1 = (r1 < nrows) ? W[(long long)idx[r1] * D + g * 16 + col] : 0.f;
            op[g] = (v2f){a0, a1};
        }
        gram_mma(c, op);
    }

    gram_reduce(c, lds, acc, col, khalf);
}

// pos_data_loss: one block per batch row b; ~210 MB of 256B row gathers.
__global__ void pos_loss_kernel(const int*   __restrict__ uids,
                                const int*   __restrict__ pos_iids,
                                const int*   __restrict__ n_item_p,
                                const float* __restrict__ userW,
                                const float* __restrict__ itemW,
                                const float* __restrict__ h,
                                int L,
                                float* __restrict__ out_accum) {
    __shared__ float uh[D];
    __shared__ float red[256];
    const int b = blockIdx.x;
    const int t = threadIdx.x;
    const int n_item = *n_item_p;

    if (t < D) {
        long long u = uids[b];
        uh[t] = userW[u * D + t] * h[t];
    }
    __syncthreads();

    float contrib = 0.f;
    for (int l = t; l < L; l += blockDim.x) {
        int iid = pos_iids[(long long)b * L + l];
        if (iid != n_item) {
            const float4* row = (const float4*)(itemW + (long long)iid * D);
            float hpq = 0.f;
#pragma unroll
            for (int q = 0; q < D / 4; ++q) {
                float4 v = row[q];
                hpq += v.x * uh[q * 4 + 0] + v.y * uh[q * 4 + 1] +
                       v.z * uh[q * 4 + 2] + v.w * uh[q * 4 + 3];
            }
            contrib += (1.f - NEG_WEIGHT) * hpq * hpq - 2.f * hpq;
        }
    }
    red[t] = contrib;
    __syncthreads();
    for (int s = blockDim.x >> 1; s > 0; s >>= 1) {
        if (t < s) red[t] += red[t + s];
        __syncthreads();
    }
    if (t == 0) atomicAdd(out_accum, red[0]);
}

__global__ void finalize_kernel(const float* __restrict__ part1,
                                const float* __restrict__ part2,
                                const float* __restrict__ h,
                                const float* __restrict__ posLoss,
                                float* __restrict__ out) {
    __shared__ float red[256];
    const int t = threadIdx.x;
    float s = 0.f;
    for (int idx = t; idx < D * D; idx += 256) {
        int i = idx >> 6, j = idx & 63;
        s += part1[idx] * part2[idx] * h[i] * h[j];
    }
    red[t] = s;
    __syncthreads();
    for (int k = 128; k > 0; k >>= 1) {
        if (t < k) red[t] += red[t + k];
        __syncthreads();
    }
    if (t == 0) out[0] = NEG_WEIGHT * red[0] + posLoss[0];
}

extern "C" void kernel_launch(void* const* d_in, const int* in_sizes, int n_in,
                              void* d_out, int out_size, void* d_ws, size_t ws_size,
                              hipStream_t stream) {
    const int*   uids     = (const int*)d_in[0];
    const int*   pos_iids = (const int*)d_in[1];
    const int*   n_item_p = (const int*)d_in[2];
    const float* userW    = (const float*)d_in[3];
    const float* itemW    = (const float*)d_in[4];
    const float* h        = (const float*)d_in[5];

    const int B        = in_sizes[0];
    const int L        = in_sizes[1] / B;
    const int itemRows = in_sizes[4] / D;   // N_ITEM + 1

    float* ws      = (float*)d_ws;
    float* part1   = ws;                    // 64*64
    float* part2   = ws + D * D;            // 64*64
    float* posLoss = ws + 2 * D * D;        // 1

    const int nZero = 2 * D * D + 1;
    init_ws_kernel<<<(nZero + 255) / 256, 256, 0, stream>>>(ws, nZero);

    // item Gram: streams 256 MB once; bandwidth-bound (~11 us at 23.3 TB/s)
    gram64_item_kernel<<<1280, GRAM_BLOCK, 0, stream>>>(itemW, itemRows, part1);
    // user Gram: 4096 gathered rows, tiny
    gram64_user_kernel<<<64, GRAM_BLOCK, 0, stream>>>(userW, uids, B, part2);
    // positive-term gather/dot
    pos_loss_kernel<<<B, 256, 0, stream>>>(uids, pos_iids, n_item_p, userW, itemW,
                                           h, L, posLoss);
    finalize_kernel<<<1, 256, 0, stream>>>(part1, part2, h, posLoss, (float*)d_out);
}